// ComnetModel_34694745817758
// MI455X (gfx1250) — compile-verified
//
#include <hip/hip_runtime.h>
#include <cstdint>

#define NLINKS 20000
#define T_ITERS 3

typedef __attribute__((ext_vector_type(2))) float v2f;
typedef __attribute__((ext_vector_type(8))) float v8f;

__device__ __forceinline__ float sigmoidf_(float x) {
    return 1.0f / (1.0f + __expf(-x));
}
__device__ __forceinline__ float tanhf_(float x) {
    x = fminf(fmaxf(x, -15.0f), 15.0f);
    float e = __expf(2.0f * x);
    return (e - 1.0f) / (e + 1.0f);
}
// Branchless SELU: one v_exp_f32, no exec-mask branches.
// x>0: max=x, exp(0)-1=0 -> scale*x ; x<=0: max=0 -> scale*alpha*(exp(x)-1)
__device__ __forceinline__ float seluf_(float x) {
    const float scale = 1.0507009873554805f;
    const float alpha = 1.6732632423543772f;
    float pos = fmaxf(x, 0.0f);
    float neg = alpha * (__expf(fminf(x, 0.0f)) - 1.0f);
    return scale * (pos + neg);
}

// ---------------------------------------------------------------- init
__global__ void init_ps_kernel(const float* __restrict__ traffic,
                               float* __restrict__ ps, int n_paths) {
    int p = blockIdx.x * blockDim.x + threadIdx.x;
    if (p < n_paths) { ps[2 * p] = traffic[p]; ps[2 * p + 1] = 0.0f; }
}
__global__ void zero_kernel(float* __restrict__ p, int n) {
    int i = blockIdx.x * blockDim.x + threadIdx.x;
    if (i < n) p[i] = 0.0f;
}

// ------------------------------------------------------- path GRU step
__device__ __forceinline__ void path_gru_step(
    int li, const float4* __restrict__ ls4,
    const float* GK, const float* GB, const float* CK, const float* CB,
    float& h0, float& h1, float* s_m)
{
    float4 x = ls4[li];  // gather link state (L2-hot 320KB table)
    float g[4];
#pragma unroll
    for (int j = 0; j < 4; ++j)
        g[j] = sigmoidf_(GB[j]
                         + x.x * GK[0 * 4 + j] + x.y * GK[1 * 4 + j]
                         + x.z * GK[2 * 4 + j] + x.w * GK[3 * 4 + j]
                         + h0  * GK[4 * 4 + j] + h1  * GK[5 * 4 + j]);
    float rh0 = g[0] * h0, rh1 = g[1] * h1;
    float c0 = tanhf_(CB[0] + x.x * CK[0] + x.y * CK[2] + x.z * CK[4]
                            + x.w * CK[6] + rh0 * CK[8] + rh1 * CK[10]);
    float c1 = tanhf_(CB[1] + x.x * CK[1] + x.y * CK[3] + x.z * CK[5]
                            + x.w * CK[7] + rh0 * CK[9] + rh1 * CK[11]);
    h0 = g[2] * h0 + (1.0f - g[2]) * c0;
    h1 = g[3] * h1 + (1.0f - g[3]) * c1;
    // per-hop output scatter-add into workgroup-local LDS copy of m
    atomicAdd(&s_m[li * 2 + 0], h0);
    atomicAdd(&s_m[li * 2 + 1], h1);
}

// ------------------------------------------------- per-path sequence GRU
__global__ void __launch_bounds__(512)
path_kernel(const int* __restrict__ links,
            const float* __restrict__ gk, const float* __restrict__ gb,
            const float* __restrict__ ck, const float* __restrict__ cb,
            const float* __restrict__ ls, float* __restrict__ ps,
            float* __restrict__ m_glob, int n_paths, int L)
{
    // full m accumulator in LDS: 20000*2*4 = 160000 bytes (CDNA5: 320KB/WGP)
    __shared__ float s_m[NLINKS * 2];
    for (int i = threadIdx.x; i < NLINKS * 2; i += blockDim.x) s_m[i] = 0.0f;

    float GK[24], GB[4], CK[12], CB[2];
#pragma unroll
    for (int i = 0; i < 24; ++i) GK[i] = gk[i];
#pragma unroll
    for (int i = 0; i < 4; ++i) GB[i] = gb[i];
#pragma unroll
    for (int i = 0; i < 12; ++i) CK[i] = ck[i];
#pragma unroll
    for (int i = 0; i < 2; ++i) CB[i] = cb[i];
    __syncthreads();

    const float4* ls4 = (const float4*)ls;
    int chunk = (n_paths + (int)gridDim.x - 1) / (int)gridDim.x;
    int p0 = (int)blockIdx.x * chunk;
    int p1 = min(p0 + chunk, n_paths);

    for (int p = p0 + (int)threadIdx.x; p < p1; p += (int)blockDim.x) {
        float h0 = ps[2 * p + 0], h1 = ps[2 * p + 1];
        const long base = (long)p * L;
        if (L == 8) {
            const int4* lp = (const int4*)(links + base);  // 32B-aligned
            int4 la = lp[0], lb = lp[1];
            int li[8] = {la.x, la.y, la.z, la.w, lb.x, lb.y, lb.z, lb.w};
#pragma unroll
            for (int k = 0; k < 8; ++k)
                path_gru_step(li[k], ls4, GK, GB, CK, CB, h0, h1, s_m);
        } else {
            for (int k = 0; k < L; ++k)
                path_gru_step(links[base + k], ls4, GK, GB, CK, CB, h0, h1, s_m);
        }
        ps[2 * p + 0] = h0; ps[2 * p + 1] = h1;
    }
    __syncthreads();
    // coalesced flush of the local partial sums
    for (int i = threadIdx.x; i < NLINKS * 2; i += blockDim.x) {
        float v = s_m[i];
        if (v != 0.0f) unsafeAtomicAdd(&m_glob[i], v);
    }
}

// ---------------------------------------------------------- edge GRU
__global__ void __launch_bounds__(256)
edge_kernel(const float* __restrict__ m,
            const float* __restrict__ gk, const float* __restrict__ gb,
            const float* __restrict__ ck, const float* __restrict__ cb,
            float* __restrict__ ls)
{
    int li = blockIdx.x * blockDim.x + threadIdx.x;
    if (li >= NLINKS) return;
    float x0 = m[2 * li], x1 = m[2 * li + 1];
    float4 hv = ((const float4*)ls)[li];
    float h[4] = {hv.x, hv.y, hv.z, hv.w};
    float g[8];
#pragma unroll
    for (int j = 0; j < 8; ++j) {
        float a = gb[j] + x0 * gk[j] + x1 * gk[8 + j];
#pragma unroll
        for (int i = 0; i < 4; ++i) a += h[i] * gk[(2 + i) * 8 + j];
        g[j] = sigmoidf_(a);
    }
    float c[4];
#pragma unroll
    for (int j = 0; j < 4; ++j) {
        float a = cb[j] + x0 * ck[j] + x1 * ck[4 + j];
#pragma unroll
        for (int i = 0; i < 4; ++i) a += (g[i] * h[i]) * ck[(2 + i) * 4 + j];
        c[j] = tanhf_(a);
    }
    float4 o;
    o.x = g[4] * h[0] + (1.0f - g[4]) * c[0];
    o.y = g[5] * h[1] + (1.0f - g[5]) * c[1];
    o.z = g[6] * h[2] + (1.0f - g[6]) * c[2];
    o.w = g[7] * h[3] + (1.0f - g[7]) * c[3];
    ((float4*)ls)[li] = o;
}

// ------------------------------------------------ readout MLP (WMMA f32)
// One wave handles 16 paths. Layer1 (2->8) is computed scalar (each lane
// pair redundantly builds one row of h1 in the ISA A-matrix 16x4 layout).
// Layer2 h2 = h1 @ w2 (16x8 @ 8x8) = two chained V_WMMA_F32_16X16X4_F32.
// Layer3 (8->1) is a 16-lane xor-shuffle dot with w3 on the C/D layout.
__global__ void __launch_bounds__(256)
readout_kernel(const float* __restrict__ ps,
               const float* __restrict__ w1, const float* __restrict__ b1,
               const float* __restrict__ w2, const float* __restrict__ b2,
               const float* __restrict__ w3, const float* __restrict__ b3,
               float* __restrict__ out, int n_paths)
{
    int lane = threadIdx.x & 31;
    int wave = (int)blockIdx.x * (int)(blockDim.x >> 5) + ((int)threadIdx.x >> 5);
    int base = wave * 16;
    if (base >= n_paths) return;          // wave-uniform: EXEC stays all-ones
    int row = lane & 15;                  // tile row (M) this lane feeds
    bool hi = lane >= 16;
    int p = min(base + row, n_paths - 1); // tail-safe (stores are guarded)

    float s0 = ps[2 * p], s1 = ps[2 * p + 1];
    float hA[8];
#pragma unroll
    for (int j = 0; j < 8; ++j)
        hA[j] = seluf_(b1[j] + s0 * w1[j] + s1 * w1[8 + j]);

    // Columns >= 8 of the 16-wide tile are padded with zeros; use clamped
    // index + mask multiply so loads are unconditional (no exec branches).
    int col  = row;
    int colm = col & 7;
    float cmask = (col < 8) ? 1.0f : 0.0f;
    float w2c[8];
#pragma unroll
    for (int k = 0; k < 8; ++k) w2c[k] = w2[k * 8 + colm] * cmask;

    // A 16x4 layout: lanes 0-15 hold K=0,1; lanes 16-31 hold K=2,3
    v2f a1, a2, bm1, bm2;
    a1.x = hi ? hA[2] : hA[0];  a1.y = hi ? hA[3] : hA[1];
    a2.x = hi ? hA[6] : hA[4];  a2.y = hi ? hA[7] : hA[5];
    // B 4x16 layout (row striped across lanes): lanes 0-15 K=0,1; 16-31 K=2,3
    bm1.x = hi ? w2c[2] : w2c[0];  bm1.y = hi ? w2c[3] : w2c[1];
    bm2.x = hi ? w2c[6] : w2c[4];  bm2.y = hi ? w2c[7] : w2c[5];

    v8f c = {0.f, 0.f, 0.f, 0.f, 0.f, 0.f, 0.f, 0.f};
    c = __builtin_amdgcn_wmma_f32_16x16x4_f32(false, a1, false, bm1,
                                              (short)0, c, false, false);
    c = __builtin_amdgcn_wmma_f32_16x16x4_f32(false, a2, false, bm2,
                                              (short)0, c, false, false);

    float b2c = b2[colm] * cmask;
    float w3c = w3[colm] * cmask;
    float acc[8];
#pragma unroll
    for (int r = 0; r < 8; ++r) acc[r] = seluf_(c[r] + b2c) * w3c;
    // sum over N (16 lanes per half); xor masks 1..8 stay within each half
#pragma unroll
    for (int msk = 1; msk < 16; msk <<= 1) {
#pragma unroll
        for (int r = 0; r < 8; ++r) acc[r] += __shfl_xor(acc[r], msk, 32);
    }
    float bout = b3[0];
    if (lane == 0) {
#pragma unroll
        for (int r = 0; r < 8; ++r) {
            int pp = base + r;
            if (pp < n_paths) out[pp] = acc[r] + bout;   // rows M=0..7
        }
    } else if (lane == 16) {
#pragma unroll
        for (int r = 0; r < 8; ++r) {
            int pp = base + 8 + r;
            if (pp < n_paths) out[pp] = acc[r] + bout;   // rows M=8..15
        }
    }
}

// ---------------------------------------------------------------- host
extern "C" void kernel_launch(void* const* d_in, const int* in_sizes, int n_in,
                              void* d_out, int out_size, void* d_ws, size_t ws_size,
                              hipStream_t stream)
{
    const float* traffic = (const float*)d_in[0];
    const float* path_gk = (const float*)d_in[1];
    const float* path_gb = (const float*)d_in[2];
    const float* path_ck = (const float*)d_in[3];
    const float* path_cb = (const float*)d_in[4];
    const float* edge_gk = (const float*)d_in[5];
    const float* edge_gb = (const float*)d_in[6];
    const float* edge_ck = (const float*)d_in[7];
    const float* edge_cb = (const float*)d_in[8];
    const float* w1 = (const float*)d_in[9];
    const float* b1 = (const float*)d_in[10];
    const float* w2 = (const float*)d_in[11];
    const float* b2 = (const float*)d_in[12];
    const float* w3 = (const float*)d_in[13];
    const float* b3 = (const float*)d_in[14];
    const int* links = (const int*)d_in[15];

    int n_paths = in_sizes[0];
    int n_total = in_sizes[15];
    int L = n_total / n_paths;

    // workspace: path_state [n_paths,2] | link_state [NLINKS,4] | m [NLINKS,2]
    float* ws = (float*)d_ws;
    size_t ps_n   = (size_t)n_paths * 2;
    size_t ls_off = (ps_n + 3) & ~(size_t)3;       // 16B align for float4
    size_t m_off  = ls_off + (size_t)NLINKS * 4;
    float* ps = ws;
    float* ls = ws + ls_off;
    float* m  = ws + m_off;
    (void)n_in; (void)out_size; (void)ws_size;

    init_ps_kernel<<<(n_paths + 255) / 256, 256, 0, stream>>>(traffic, ps, n_paths);
    zero_kernel<<<(NLINKS * 4 + 255) / 256, 256, 0, stream>>>(ls, NLINKS * 4);

    for (int t = 0; t < T_ITERS; ++t) {
        zero_kernel<<<(NLINKS * 2 + 255) / 256, 256, 0, stream>>>(m, NLINKS * 2);
        path_kernel<<<256, 512, 0, stream>>>(links, path_gk, path_gb, path_ck,
                                             path_cb, ls, ps, m, n_paths, L);
        edge_kernel<<<(NLINKS + 255) / 256, 256, 0, stream>>>(
            m, edge_gk, edge_gb, edge_ck, edge_cb, ls);
    }

    int ntiles = (n_paths + 15) / 16;
    int wpb = 256 / 32;
    int nblocks = (ntiles + wpb - 1) / wpb;
    readout_kernel<<<nblocks, 256, 0, stream>>>(ps, w1, b1, w2, b2, w3, b3,
                                                (float*)d_out, n_paths);
}